// OKF_7584912245094
// MI455X (gfx1250) — compile-verified
//
#include <hip/hip_runtime.h>
#include <hip/hip_bf16.h>

// CDNA5 wave32 WMMA operand vectors
typedef __attribute__((ext_vector_type(2))) float v2f;   // 16x4 f32 A-operand / 4x16 B-operand
typedef __attribute__((ext_vector_type(8))) float v8f;   // 16x16 f32 C/D

#define DX 8
#define DZ 4
#define KF_P0 1000.0f
#define IMG_STRIDE 192   // per-step operand image: 6 "VGPRs" x 32 lanes (A chunks 0..3, K chunks 4..5)
#define WAVES 4          // waves (tiles of 16 trajectories) per block

// ---------------------------------------------------------------------------
// Phase 1: data-independent Riccati recursion. One workgroup computes
// P_t / S_t / K_t / A_t=(I-K_tH)F for all t, and emits A_t,K_t pre-swizzled
// into the exact VGPR layout of the 16x16x4 f32 WMMA A-operand
// (lane m<16: M=m,K=k ; lanes 16-31: M=m,K=k+2 ; rows 8..15 zero).
// ---------------------------------------------------------------------------
__global__ __launch_bounds__(64) void okf_riccati(
    const float* __restrict__ Fg, const float* __restrict__ Hg,
    const float* __restrict__ Q_D, const float* __restrict__ Q_L,
    const float* __restrict__ R_D, const float* __restrict__ R_L,
    float* __restrict__ img, int T)
{
  __shared__ float Fs[64], Hs[32], Q[64], R[16], Aq[64], Ar[16];
  __shared__ float P[64], Pp[64], PHt[32], S[16], Sinv[16];
  __shared__ float K[32], IKH[64], A[64], KR[32], T1[64], M[64];

  const int tid = threadIdx.x;
  const int i = tid >> 3, j = tid & 7;

  Fs[tid] = Fg[tid];
  if (tid < 32) Hs[tid] = Hg[tid];
  { // Cholesky-factor parameterization: Aq = diag(exp(Q_D)) + strict-lower(Q_L)
    float v = 0.f;
    if (i == j)      v = expf(Q_D[i]);
    else if (i > j)  v = Q_L[(i * (i - 1)) / 2 + j];   // tril_indices(-1) row-major
    Aq[tid] = v;
  }
  if (tid < 16) {
    const int ri = tid >> 2, rj = tid & 3;
    float v = 0.f;
    if (ri == rj)     v = expf(R_D[ri]);
    else if (ri > rj) v = R_L[(ri * (ri - 1)) / 2 + rj];
    Ar[tid] = v;
  }
  __syncthreads();
  { float s = 0.f; for (int k = 0; k < 8; ++k) s += Aq[i*8+k] * Aq[j*8+k]; Q[tid] = s; }
  if (tid < 16) {
    const int ri = tid >> 2, rj = tid & 3;
    float s = 0.f; for (int k = 0; k < 4; ++k) s += Ar[ri*4+k] * Ar[rj*4+k]; R[tid] = s;
  }
  Pp[tid] = (i == j) ? KF_P0 : 0.f;   // first update runs on P0*I (no predict)
  __syncthreads();

  for (int t = 0; t < T; ++t) {
    if (t > 0) { // predict: Pp = F P F^T + Q
      { float s = 0.f; for (int k = 0; k < 8; ++k) s += Fs[i*8+k] * P[k*8+j]; T1[tid] = s; }
      __syncthreads();
      { float s = Q[tid]; for (int k = 0; k < 8; ++k) s += T1[i*8+k] * Fs[j*8+k]; Pp[tid] = s; }
      __syncthreads();
    }
    // PHt = Pp H^T (8x4)
    if (tid < 32) {
      const int r = tid >> 2, cc = tid & 3;
      float s = 0.f; for (int k = 0; k < 8; ++k) s += Pp[r*8+k] * Hs[cc*8+k];
      PHt[r*4+cc] = s;
    }
    __syncthreads();
    // S = H PHt + R (4x4)
    if (tid < 16) {
      const int cc = tid >> 2, dd = tid & 3;
      float s = R[cc*4+dd]; for (int k = 0; k < 8; ++k) s += Hs[cc*8+k] * PHt[k*4+dd];
      S[cc*4+dd] = s;
    }
    __syncthreads();
    // 4x4 inverse (Gauss-Jordan, partial pivot) by one lane
    if (tid == 0) {
      float a[4][8];
      for (int r = 0; r < 4; ++r)
        for (int cc = 0; cc < 4; ++cc) { a[r][cc] = S[r*4+cc]; a[r][4+cc] = (r == cc) ? 1.f : 0.f; }
      for (int col = 0; col < 4; ++col) {
        int piv = col; float mx = fabsf(a[col][col]);
        for (int r = col + 1; r < 4; ++r) { float v = fabsf(a[r][col]); if (v > mx) { mx = v; piv = r; } }
        if (piv != col) for (int cc = 0; cc < 8; ++cc) { float tv = a[col][cc]; a[col][cc] = a[piv][cc]; a[piv][cc] = tv; }
        const float inv = 1.0f / a[col][col];
        for (int cc = 0; cc < 8; ++cc) a[col][cc] *= inv;
        for (int r = 0; r < 4; ++r) if (r != col) {
          const float f = a[r][col];
          for (int cc = 0; cc < 8; ++cc) a[r][cc] -= f * a[col][cc];
        }
      }
      for (int r = 0; r < 4; ++r) for (int cc = 0; cc < 4; ++cc) Sinv[r*4+cc] = a[r][4+cc];
    }
    __syncthreads();
    // K = PHt Sinv (8x4)
    if (tid < 32) {
      const int r = tid >> 2, cc = tid & 3;
      float s = 0.f; for (int dd = 0; dd < 4; ++dd) s += PHt[r*4+dd] * Sinv[dd*4+cc];
      K[r*4+cc] = s;
    }
    __syncthreads();
    // IKH = I - K H
    { float s = (i == j) ? 1.f : 0.f;
      for (int cc = 0; cc < 4; ++cc) s -= K[i*4+cc] * Hs[cc*8+j];
      IKH[tid] = s; }
    __syncthreads();
    // A = IKH F ; KR = K R ; T1 = IKH Pp
    { float s = 0.f; for (int k = 0; k < 8; ++k) s += IKH[i*8+k] * Fs[k*8+j]; A[tid] = s; }
    if (tid < 32) {
      const int r = tid >> 2, dd = tid & 3;
      float s = 0.f; for (int cc = 0; cc < 4; ++cc) s += K[r*4+cc] * R[cc*4+dd];
      KR[r*4+dd] = s;
    }
    { float s = 0.f; for (int k = 0; k < 8; ++k) s += IKH[i*8+k] * Pp[k*8+j]; T1[tid] = s; }
    __syncthreads();
    // Joseph: M = T1 IKH^T + KR K^T ; P = sym(M)
    { float s = 0.f;
      for (int k = 0; k < 8; ++k)  s += T1[i*8+k] * IKH[j*8+k];
      for (int dd = 0; dd < 4; ++dd) s += KR[i*4+dd] * K[j*4+dd];
      M[tid] = s; }
    __syncthreads();
    P[tid] = 0.5f * (M[i*8+j] + M[j*8+i]);
    // Emit WMMA A-operand image for this step (t>=1 drives x_t)
    if (t > 0) {
      float* dst = img + (size_t)(t - 1) * IMG_STRIDE;
      for (int e = tid; e < IMG_STRIDE; e += 64) {
        const int v = e >> 5, l = e & 31;
        const int m = l & 15;
        const int up = (l >> 4) << 1;      // lanes 16-31 carry K+2
        float val = 0.f;
        if (m < 8) {
          if (v < 4) { const int kk = ((v >> 1) << 2) + (v & 1) + up; val = A[m*8+kk]; }
          else       { const int kk = (v & 1) + up;                   val = K[m*4+kk]; }
        }
        dst[e] = val;
      }
    }
    __syncthreads();
  }
}

// ---------------------------------------------------------------------------
// Phase 2: batched recurrence x_t = A_t x_{t-1} + K_t z_t via 16x16x4 f32 WMMA.
// One wave owns 16 trajectories. X lives in LDS in C/D layout between steps;
// Z and out are staged in 16-step chunks so every lane does contiguous
// 128b global accesses. The A_t/K_t operand image is software-pipelined one
// step ahead so its global loads are never on the serial critical path.
// ---------------------------------------------------------------------------
__global__ __launch_bounds__(WAVES * 32) void okf_scan(
    const float* __restrict__ Z, const float* __restrict__ img,
    float* __restrict__ out, int T)
{
  __shared__ float xbuf[WAVES][16][8][16];   // [wave][t%16][state][traj]
  __shared__ float zbuf[WAVES][16][4][16];   // [wave][t%16][obs][traj]

  const int lane = threadIdx.x & 31;
  const int w    = threadIdx.x >> 5;
  const int tile = blockIdx.x * WAVES + w;   // 16 trajectories per tile
  const int n    = lane & 15;                // traj within tile (WMMA N / B-layout role)
  const int half = lane >> 4;                // 0: rows k ; 1: rows k+2 (B-operand halves)
  const int tn   = lane >> 1;                // traj for staging roles
  const int hl   = lane & 1;                 // time-half for staging roles
  const size_t traj0 = (size_t)tile * 16;

  // x0 = [z0 ; 0]
  if (lane < 16) {
    const float4 z0 = *(const float4*)(Z + (traj0 + n) * (size_t)T * 4);
    xbuf[w][0][0][n] = z0.x; xbuf[w][0][1][n] = z0.y;
    xbuf[w][0][2][n] = z0.z; xbuf[w][0][3][n] = z0.w;
    xbuf[w][0][4][n] = 0.f;  xbuf[w][0][5][n] = 0.f;
    xbuf[w][0][6][n] = 0.f;  xbuf[w][0][7][n] = 0.f;
  }

  // Prime the operand pipeline with step t=1 (image slot 0).
  v2f a0, a1, kv;
  {
    const float* ip0 = img + lane;
    a0.x = ip0[0];   a0.y = ip0[32];
    a1.x = ip0[64];  a1.y = ip0[96];
    kv.x = ip0[128]; kv.y = ip0[160];
  }

  for (int c = 0; c < T / 16; ++c) {
    const int tbase = c * 16;

    // Stage Z[traj][tbase..tbase+15][0..3]: each lane reads 128 contiguous bytes.
    {
      const float* zp = Z + ((traj0 + tn) * (size_t)T + tbase + hl * 8) * 4;
      #pragma unroll
      for (int k = 0; k < 8; ++k) {
        const float4 zv = *(const float4*)(zp + k * 4);
        const int tt = hl * 8 + k;
        zbuf[w][tt][0][tn] = zv.x; zbuf[w][tt][1][tn] = zv.y;
        zbuf[w][tt][2][tn] = zv.z; zbuf[w][tt][3][tn] = zv.w;
      }
    }

    const int tstart = (c == 0) ? 1 : tbase;
    for (int t = tstart; t < tbase + 16; ++t) {
      // Prefetch A_{t+1}/K_{t+1} operand image (clamped to the last valid slot);
      // these 6 coalesced b32 loads overlap the WMMA/LDS chain below.
      const int tpf = (t + 1 < T) ? t + 1 : T - 1;
      const float* ipn = img + (size_t)(tpf - 1) * IMG_STRIDE + lane;
      v2f na0, na1, nkv;
      na0.x = ipn[0];   na0.y = ipn[32];
      na1.x = ipn[64];  na1.y = ipn[96];
      nkv.x = ipn[128]; nkv.y = ipn[160];

      // B-operand (4x16 f32): V0 = [row k | row k+2], V1 = [row k+1 | row k+3]
      const int ps = (t - 1) & 15;
      v2f b0, b1, bz;
      b0.x = xbuf[w][ps][2 * half + 0][n];  b0.y = xbuf[w][ps][2 * half + 1][n];
      b1.x = xbuf[w][ps][2 * half + 4][n];  b1.y = xbuf[w][ps][2 * half + 5][n];
      const int tm = t & 15;
      bz.x = zbuf[w][tm][2 * half + 0][n];  bz.y = zbuf[w][tm][2 * half + 1][n];

      // x_t = A_t x_{t-1} + K_t z_t  (3 chained f32 WMMAs, K=4 each)
      v8f d = {};
      d = __builtin_amdgcn_wmma_f32_16x16x4_f32(false, a0, false, b0, (short)0, d, false, false);
      d = __builtin_amdgcn_wmma_f32_16x16x4_f32(false, a1, false, b1, (short)0, d, false, false);
      d = __builtin_amdgcn_wmma_f32_16x16x4_f32(false, kv, false, bz, (short)0, d, false, false);

      // D layout: VGPR s, lanes 0-15 = (state s, traj lane); rows 8-15 are exact zeros.
      if (lane < 16) {
        #pragma unroll
        for (int s = 0; s < 8; ++s) xbuf[w][tm][s][lane] = d[s];
      }

      a0 = na0; a1 = na1; kv = nkv;   // rotate pipeline
    }

    // Flush chunk: each lane writes 256 contiguous bytes of one trajectory.
    {
      float* op = out + ((traj0 + tn) * (size_t)T + tbase + hl * 8) * 8;
      #pragma unroll
      for (int k = 0; k < 8; ++k) {
        const int tt = hl * 8 + k;
        float4 lo, hi;
        lo.x = xbuf[w][tt][0][tn]; lo.y = xbuf[w][tt][1][tn];
        lo.z = xbuf[w][tt][2][tn]; lo.w = xbuf[w][tt][3][tn];
        hi.x = xbuf[w][tt][4][tn]; hi.y = xbuf[w][tt][5][tn];
        hi.z = xbuf[w][tt][6][tn]; hi.w = xbuf[w][tt][7][tn];
        *(float4*)(op + (size_t)k * 8)     = lo;
        *(float4*)(op + (size_t)k * 8 + 4) = hi;
      }
    }
  }
}

extern "C" void kernel_launch(void* const* d_in, const int* in_sizes, int n_in,
                              void* d_out, int out_size, void* d_ws, size_t ws_size,
                              hipStream_t stream) {
  (void)n_in; (void)out_size; (void)ws_size;
  const float* Z   = (const float*)d_in[0];
  const float* F   = (const float*)d_in[1];
  const float* H   = (const float*)d_in[2];
  const float* Q_D = (const float*)d_in[3];
  const float* Q_L = (const float*)d_in[4];
  const float* R_D = (const float*)d_in[5];
  const float* R_L = (const float*)d_in[6];
  float* out = (float*)d_out;
  float* img = (float*)d_ws;            // (T-1)*192 floats ≈ 768 KB of scratch

  const int T = 1024;
  const int B = in_sizes[0] / (T * DZ); // 4096

  okf_riccati<<<1, 64, 0, stream>>>(F, H, Q_D, Q_L, R_D, R_L, img, T);

  const int tiles = B / 16;             // 256 waves total
  okf_scan<<<tiles / WAVES, WAVES * 32, 0, stream>>>(Z, img, out, T);
}